// AtBatCell_15977278341980
// MI455X (gfx1250) — compile-verified
//
#include <hip/hip_runtime.h>
#include <math.h>

typedef float v2f __attribute__((ext_vector_type(2)));
typedef float v8f __attribute__((ext_vector_type(8)));
typedef unsigned int v4u __attribute__((ext_vector_type(4)));
typedef int v4i __attribute__((ext_vector_type(4)));
typedef int v8i __attribute__((ext_vector_type(8)));

#define AS1 __attribute__((address_space(1)))
#define AS3 __attribute__((address_space(3)))

#define T_STEPS   8192
#define SITD      64
#define SROW      256
#define S2D       512
#define NPLAYERS  32000
#define GBLK      32          // persistent blocks; each owns 512/32 = 16 output rows
#define RPB       16          // rows per block
#define UROW      544         // 512 + 2*16 pad (16 DWORDs after each 256-DWORD chunk)

// CDNA5 feature detection (compile-only environment; keep safe fallbacks)
#if defined(__AMDGCN__) && __has_builtin(__builtin_amdgcn_global_load_async_to_lds_b128)
#define HAVE_ASYNC_LDS 1
#else
#define HAVE_ASYNC_LDS 0
#endif
#if defined(__AMDGCN__) && __has_builtin(__builtin_amdgcn_tensor_load_to_lds)
#define HAVE_TDM 1
#else
#define HAVE_TDM 0
#endif
#if __has_include(<hip/amd_detail/amd_gfx1250_TDM.h>)
#define TDM_ARITY6 1
#else
#define TDM_ARITY6 0
#endif

#define CPOL_SCOPE_DEV 0x10   // CPol: scope bits [4:3] = 2 (DEV)

// workspace layout (bytes)
#define WS_BAR    0           // unsigned barrier counter
#define WS_RVEC   64          // 512 f32 r-gate broadcast vector
#define WS_X      4096        // Xz | Xr | Xh, each T_STEPS*S2D f32 (16 MB each)

// LDS layout for a U row slice: 16 DWORDs of pad after every 256 DWORDs.
// addr(r, k) = r*544 + k + 16*(k >= 256); rows r and r+1 differ by 544
// (544 % 64 == 32), so the two half-wave lane groups hit disjoint bank sets.
__device__ __forceinline__ int uidx(int r, int k) {
    return r * UROW + k + ((k >> 8) << 4);
}

__device__ __forceinline__ void wait_asynccnt0() {
#if defined(__AMDGCN__)
#if __has_builtin(__builtin_amdgcn_s_wait_asynccnt)
    __builtin_amdgcn_s_wait_asynccnt(0);
#elif HAVE_ASYNC_LDS
    asm volatile("s_wait_asynccnt 0x0" ::: "memory");
#endif
#endif
}

__device__ __forceinline__ void wait_tensorcnt0() {
#if defined(__AMDGCN__)
#if __has_builtin(__builtin_amdgcn_s_wait_tensorcnt)
    __builtin_amdgcn_s_wait_tensorcnt(0);
#elif HAVE_TDM
    asm volatile("s_wait_tensorcnt 0x0" ::: "memory");
#endif
#endif
}

// ---- workgroup-cluster support (graceful when not dispatched in clusters) ----
// Physical cluster id: IB_STS2[9:6]; 0 = not in a cluster.
__device__ __forceinline__ unsigned phys_cluster_id() {
#if defined(__AMDGCN__)
    unsigned v;
    asm volatile("s_getreg_b32 %0, hwreg(HW_REG_IB_STS2, 6, 4)" : "=s"(v));
    return v;
#else
    return 0;
#endif
}
// Logical workgroup id within cluster: IB_STS2[24:21]; 0 when not in cluster.
__device__ __forceinline__ unsigned wg_in_cluster() {
#if defined(__AMDGCN__)
    unsigned v;
    asm volatile("s_getreg_b32 %0, hwreg(HW_REG_IB_STS2, 21, 4)" : "=s"(v));
    return v;
#else
    return 0;
#endif
}
// Workgroups per cluster: TTMP6[27:24] = flat_NWG - 1; TTMP6 is 0 off-cluster.
__device__ __forceinline__ unsigned cluster_nwg() {
#if defined(__AMDGCN__)
    unsigned v;
    asm volatile("s_mov_b32 %0, ttmp6" : "=s"(v));
    return ((v >> 24) & 0xfu) + 1u;
#else
    return 1;
#endif
}
// Cluster-wide split barrier (s_barrier_signal -3 / s_barrier_wait -3).
// NOP per ISA when Cluster_ID == 0.
__device__ __forceinline__ void cluster_barrier() {
#if defined(__AMDGCN__)
#if __has_builtin(__builtin_amdgcn_s_cluster_barrier)
    __builtin_amdgcn_s_cluster_barrier();
#else
    asm volatile("s_barrier_signal -3\n\ts_barrier_wait -3" ::: "memory");
#endif
#endif
}

// Per-lane async 16-byte global -> LDS copy (ASYNCcnt-tracked).
// Builtin signature (from hipcc diagnostics): (AS1 v4i*, AS3 v4i*, imm, cpol).
__device__ __forceinline__ void async_copy16(const float* gsrc, float* ldst,
                                             int cpol_scope) {
#if HAVE_ASYNC_LDS
    if (cpol_scope) {
        __builtin_amdgcn_global_load_async_to_lds_b128(
            (AS1 v4i*)(unsigned long long)gsrc,
            (AS3 v4i*)(unsigned long long)ldst, 0, CPOL_SCOPE_DEV);
    } else {
        __builtin_amdgcn_global_load_async_to_lds_b128(
            (AS1 v4i*)(unsigned long long)gsrc,
            (AS3 v4i*)(unsigned long long)ldst, 0, 0);
    }
#else
    (void)gsrc; (void)ldst; (void)cpol_scope;
#endif
}

// ---------------------------------------------------------------------------
// Init: zero the grid-barrier counter and the r broadcast vector.
// ---------------------------------------------------------------------------
__global__ void init_ws(unsigned* bar, float* rvec) {
    if (threadIdx.x == 0) *bar = 0u;
    for (int i = threadIdx.x; i < S2D; i += blockDim.x) rvec[i] = 0.0f;
}

// ---------------------------------------------------------------------------
// Gate GEMM: Xg[t][j] = sum_k x[t][k] * Wg[j][k], for g in {z,r,h}.
// One 16x16 output tile per wave via V_WMMA_F32_16X16X4_F32 (exact f32).
// ---------------------------------------------------------------------------
__global__ __launch_bounds__(256) void gate_gemm(
    const float* __restrict__ x,
    const float* __restrict__ Wz, const float* __restrict__ Wr,
    const float* __restrict__ Wh,
    float* __restrict__ Xout)
{
    const int wave  = blockIdx.x * (blockDim.x >> 5) + (threadIdx.x >> 5);
    const int lane  = threadIdx.x & 31;
    const int tilesJ = S2D / 16;                 // 32
    const int tilesT = T_STEPS / 16;             // 512
    const int perGate = tilesT * tilesJ;         // 16384
    const int gate = wave / perGate;
    if (gate >= 3) return;
    const int rem = wave % perGate;
    const int t0  = (rem / tilesJ) * 16;
    const int j0  = (rem % tilesJ) * 16;

    const float* __restrict__ W = (gate == 0) ? Wz : (gate == 1) ? Wr : Wh;
    float* __restrict__ Xg = Xout + (size_t)gate * T_STEPS * S2D;

    const int hh = lane >> 4;     // half-wave select
    const int mm = lane & 15;

    v8f c = {};
#pragma unroll
    for (int k0 = 0; k0 < SITD; k0 += 4) {
        v2f a, b;
        const int ka = k0 + 2 * hh;
        a[0] = x[(size_t)(t0 + mm) * SITD + ka + 0];
        a[1] = x[(size_t)(t0 + mm) * SITD + ka + 1];
        b[0] = W[(size_t)(j0 + mm) * SITD + ka + 0];
        b[1] = W[(size_t)(j0 + mm) * SITD + ka + 1];
        c = __builtin_amdgcn_wmma_f32_16x16x4_f32(false, a, false, b,
                                                  (short)0, c, false, false);
    }
#pragma unroll
    for (int i = 0; i < 8; ++i) {
        Xg[(size_t)(t0 + i + 8 * hh) * S2D + (j0 + mm)] = c[i];
    }
}

// ---------------------------------------------------------------------------
// Hierarchical grid barrier:
//   1) cluster barrier (all WGs of my cluster arrive; NOP off-cluster)
//   2) one rep WG per cluster bumps the monotonic global counter and spins
//   3) cluster barrier (release)
// Off-cluster this degenerates to the plain atomic barrier (every WG is rep).
// ---------------------------------------------------------------------------
__device__ __forceinline__ void grid_barrier(unsigned* bar, unsigned* epoch,
                                             unsigned nreps, bool is_rep) {
    __threadfence();
    __syncthreads();
    if (threadIdx.x == 0) {
        cluster_barrier();
        if (is_rep) {
            atomicAdd(bar, 1u);
            const unsigned want = (++(*epoch)) * nreps;
            while (__hip_atomic_load(bar, __ATOMIC_RELAXED,
                                     __HIP_MEMORY_SCOPE_AGENT) < want) {
                __builtin_amdgcn_s_sleep(2);
            }
        } else {
            ++(*epoch);
        }
        cluster_barrier();
    }
    __syncthreads();
}

// ---------------------------------------------------------------------------
// Persistent recurrent kernel. 32 blocks x 256 threads; block g owns output
// rows j in [g*16, g*16+16). U rows live in LDS for all 8192 steps.
// ---------------------------------------------------------------------------
__global__ __launch_bounds__(256, 1) void recurrent(
    float* __restrict__ table,                 // d_out, NPLAYERS x SROW
    const int*   __restrict__ b,  const int*   __restrict__ p,
    const float* __restrict__ Uz, const float* __restrict__ Ur,
    const float* __restrict__ Uh,
    const float* __restrict__ bz, const float* __restrict__ br,
    const float* __restrict__ bh,
    const float* __restrict__ Xz, const float* __restrict__ Xr,
    const float* __restrict__ Xh,
    unsigned* __restrict__ bar, float* __restrict__ rvec)
{
    __shared__ float Ul[3][RPB * UROW];        // ~102 KB
    __shared__ float hsh[S2D];
    __shared__ float rtmp[S2D];
    __shared__ float rhsh[S2D];
    __shared__ float zloc[RPB];

    const int g   = blockIdx.x;
    const int tid = threadIdx.x;
    const int j0  = g * RPB;

    // Cluster topology (uniform scalars). Off-cluster: nwg=1 -> every WG is a
    // rep and nreps == gridDim.x, reproducing the plain atomic barrier.
    const unsigned nwg    = cluster_nwg();
    const bool     is_rep = (wg_in_cluster() == 0) || (phys_cluster_id() == 0);
    const unsigned nreps  = (unsigned)gridDim.x / nwg;

    // ---- stage owned U rows into LDS (once) ----
#if HAVE_TDM
    if (tid < 32) {
        for (int gate = 0; gate < 3; ++gate) {
            const float* U = (gate == 0) ? Uz : (gate == 1) ? Ur : Uh;
            const unsigned long long ga =
                (unsigned long long)&U[(size_t)j0 * S2D];
            const unsigned ldsoff =
                (unsigned)(unsigned long long)(void*)&Ul[gate][0];
            // D# group 0: count=1 | lds_addr | global_addr[56:0] | type=2
            v4u g0;
            g0.x = 1u;
            g0.y = ldsoff;
            g0.z = (unsigned)(ga & 0xffffffffull);
            g0.w = (unsigned)((ga >> 32) & 0x01ffffffull) | (2u << 30);
            // D# group 1: data_size=4B, pad 16 DW after every 256 DW,
            // tensor 512x512 f32, tile 512x16, strides 512.
            v8i g1;
            g1[0] = (int)((2u << 16) | (1u << 20) | (7u << 22) | (15u << 25));
            g1[1] = (int)(512u << 16);   // tensor_dim0 = 512
            g1[2] = (int)(512u << 16);   // tensor_dim1 = 512
            g1[3] = (int)(512u << 16);   // tile_dim0   = 512
            g1[4] = 16;                  // tile_dim1 = 16, tile_dim2 = 0
            g1[5] = 512;                 // tensor_dim0_stride = 512
            g1[6] = (int)(512u << 16);   // tensor_dim1_stride = 512
            g1[7] = 0;
            v4i g2 = {0, 0, 0, 0};
            v4i g3 = {0, 0, 0, 0};
#if TDM_ARITY6
            v8i g4 = {0, 0, 0, 0, 0, 0, 0, 0};
            __builtin_amdgcn_tensor_load_to_lds(g0, g1, g2, g3, g4, 0);
#else
            __builtin_amdgcn_tensor_load_to_lds(g0, g1, g2, g3, 0);
#endif
        }
        wait_tensorcnt0();
    }
#else
    for (int idx = tid; idx < 3 * RPB * (S2D / 4); idx += blockDim.x) {
        const int gate = idx / (RPB * (S2D / 4));
        const int rr   = (idx / (S2D / 4)) % RPB;
        const int k4   = (idx % (S2D / 4)) * 4;
        const float* __restrict__ U = (gate == 0) ? Uz : (gate == 1) ? Ur : Uh;
        const float4 v = *(const float4*)&U[(size_t)(j0 + rr) * S2D + k4];
        const int base = uidx(rr, k4);
        Ul[gate][base + 0] = v.x;
        Ul[gate][base + 1] = v.y;
        Ul[gate][base + 2] = v.z;
        Ul[gate][base + 3] = v.w;
    }
#endif
    __syncthreads();

    const int rg = tid >> 4;          // owned row within block: 0..15
    const int lg = tid & 15;          // lane within 16-lane dot group
    const int jglob = j0 + rg;        // global output index: 0..511
    const float bzj = bz[jglob];
    const float brj = br[jglob];
    const float bhj = bh[jglob];

    unsigned epoch = 0;

    for (int t = 0; t < T_STEPS; ++t) {
        const int bt = b[t];
        const int pt = p[t];

        // ---- gather h = concat(table[bt], table[pt]) into LDS.
        //      DEV scope so we see other WGPs' scatter-adds. ----
#if HAVE_ASYNC_LDS
        if (tid < 128) {
            const float* src = (tid < 64)
                ? &table[(size_t)bt * SROW + tid * 4]
                : &table[(size_t)pt * SROW + (tid - 64) * 4];
            async_copy16(src, &hsh[tid * 4], 1);
        }
        wait_asynccnt0();
        __syncthreads();
#else
        for (int i = tid; i < SROW; i += blockDim.x) {
            hsh[i] = __hip_atomic_load(&table[(size_t)bt * SROW + i],
                                       __ATOMIC_RELAXED, __HIP_MEMORY_SCOPE_AGENT);
            hsh[SROW + i] = __hip_atomic_load(&table[(size_t)pt * SROW + i],
                                       __ATOMIC_RELAXED, __HIP_MEMORY_SCOPE_AGENT);
        }
        __syncthreads();
#endif

        // ---- phase 1: z_j, r_j = sigmoid(Xg[t][j] +- bias + Ug[j]·h) ----
        float az = 0.0f, ar = 0.0f;
#pragma unroll 4
        for (int i = 0; i < 32; ++i) {
            const int k = lg + 16 * i;
            const float hk = hsh[k];
            az = fmaf(Ul[0][uidx(rg, k)], hk, az);
            ar = fmaf(Ul[1][uidx(rg, k)], hk, ar);
        }
#pragma unroll
        for (int m = 8; m >= 1; m >>= 1) {
            az += __shfl_xor(az, m);
            ar += __shfl_xor(ar, m);
        }
        if (lg == 0) {
            float zz = az + Xz[(size_t)t * S2D + jglob] + bzj;
            float rr = ar + Xr[(size_t)t * S2D + jglob] - brj;
            zz = 1.0f / (1.0f + __expf(-zz));
            rr = 1.0f / (1.0f + __expf(-rr));
            zloc[rg] = zz;
            __hip_atomic_store(&rvec[jglob], rr,
                               __ATOMIC_RELAXED, __HIP_MEMORY_SCOPE_AGENT);
            if (t + 1 < T_STEPS) {
                __builtin_prefetch(&Xz[(size_t)(t + 1) * S2D + jglob], 0, 0);
                __builtin_prefetch(&Xr[(size_t)(t + 1) * S2D + jglob], 0, 0);
                __builtin_prefetch(&Xh[(size_t)(t + 1) * S2D + jglob], 0, 0);
            }
        }
        grid_barrier(bar, &epoch, nreps, is_rep);

        // ---- rh = r * h (full vector, local copy) ----
#if HAVE_ASYNC_LDS
        if (tid < 128) {
            async_copy16(&rvec[tid * 4], &rtmp[tid * 4], 1);
        }
        wait_asynccnt0();
        __syncthreads();
        for (int i = tid; i < S2D; i += blockDim.x) {
            rhsh[i] = rtmp[i] * hsh[i];
        }
        __syncthreads();
#else
        for (int i = tid; i < S2D; i += blockDim.x) {
            const float rv = __hip_atomic_load(&rvec[i], __ATOMIC_RELAXED,
                                               __HIP_MEMORY_SCOPE_AGENT);
            rhsh[i] = rv * hsh[i];
        }
        __syncthreads();
#endif

        // ---- phase 2: m_j = tanh(...); hn_j; scatter-add delta ----
        float am = 0.0f;
#pragma unroll 4
        for (int i = 0; i < 32; ++i) {
            const int k = lg + 16 * i;
            am = fmaf(Ul[2][uidx(rg, k)], rhsh[k], am);
        }
#pragma unroll
        for (int m = 8; m >= 1; m >>= 1) am += __shfl_xor(am, m);

        if (lg == 0) {
            const float mm = tanhf(am + Xh[(size_t)t * S2D + jglob] + bhj);
            const float zz = zloc[rg];
            const float hj = hsh[jglob];
            const float hn = zz * hj + (1.0f - zz) * mm;
            const float d  = hn - hj;
            const int row  = (jglob < SROW) ? bt : pt;
            atomicAdd(&table[(size_t)row * SROW + (jglob & (SROW - 1))], d);
        }
        grid_barrier(bar, &epoch, nreps, is_rep);
    }
}

// ---------------------------------------------------------------------------
extern "C" void kernel_launch(void* const* d_in, const int* in_sizes, int n_in,
                              void* d_out, int out_size, void* d_ws, size_t ws_size,
                              hipStream_t stream) {
    const float* x      = (const float*)d_in[0];
    const int*   b      = (const int*)  d_in[1];
    const int*   p      = (const int*)  d_in[2];
    const float* Wz     = (const float*)d_in[3];
    const float* Wr     = (const float*)d_in[4];
    const float* Wh     = (const float*)d_in[5];
    const float* Uz     = (const float*)d_in[6];
    const float* Ur     = (const float*)d_in[7];
    const float* Uh     = (const float*)d_in[8];
    const float* bz     = (const float*)d_in[9];
    const float* br     = (const float*)d_in[10];
    const float* bh     = (const float*)d_in[11];
    const float* table0 = (const float*)d_in[12];

    float*    table = (float*)d_out;
    char*     ws    = (char*)d_ws;
    unsigned* bar   = (unsigned*)(ws + WS_BAR);
    float*    rvec  = (float*)(ws + WS_RVEC);
    float*    X     = (float*)(ws + WS_X);     // Xz | Xr | Xh
    const size_t Xstride = (size_t)T_STEPS * S2D;

    // table starts as table0; recurrent kernel updates it in place in d_out.
    (void)hipMemcpyAsync(table, table0, sizeof(float) * (size_t)NPLAYERS * SROW,
                         hipMemcpyDeviceToDevice, stream);

    init_ws<<<1, 256, 0, stream>>>(bar, rvec);

    // 3 gates * 512 t-tiles * 32 j-tiles = 49152 wave-tiles, 8 waves/block
    const int nblocks = (3 * (T_STEPS / 16) * (S2D / 16)) / 8;
    gate_gemm<<<nblocks, 256, 0, stream>>>(x, Wz, Wr, Wh, X);

    recurrent<<<GBLK, 256, 0, stream>>>(table, b, p, Uz, Ur, Uh, bz, br, bh,
                                        X, X + Xstride, X + 2 * Xstride,
                                        bar, rvec);
}